// GraphAttention_88553635709392
// MI455X (gfx1250) — compile-verified
//
#include <hip/hip_runtime.h>
#include <hip/hip_bf16.h>

// GAT forward for MI455X (gfx1250, wave32).
// Strategy: fold the 128x256 projection into a 128x16 "alpha" weight (proj is
// only consumed through per-head dot products), do both GEMMs with full-f32
// V_WMMA_F32_16X16X4_F32 (problem is edge-memory-bound; low precision buys
// nothing), and handle the edge phase with L2-resident gathers + f32 atomics.

typedef __attribute__((ext_vector_type(2))) float v2f;
typedef __attribute__((ext_vector_type(8))) float v8f;

#define GAT_N   50000
#define GAT_E   800000
#define GAT_HID 128
#define GAT_H   8

// -------- workspace layout (float offsets) --------
#define OFF_WALPHA 0                        // 128*16
#define OFF_BALPHA (OFF_WALPHA + 2048)      // 16
#define OFF_ALPHA  (OFF_BALPHA + 16)        // N*16
#define OFF_SCORES (OFF_ALPHA + GAT_N*16)   // E*8 (scores, then reused for exp_s)
#define OFF_SEGMAX (OFF_SCORES + GAT_E*8)   // N*8   } zeroed
#define OFF_SEGSUM (OFF_SEGMAX + GAT_N*8)   // N*8   } as one
#define OFF_AGG    (OFF_SEGSUM + GAT_N*8)   // N*128 } memset
#define ZERO_FLOATS (GAT_N*8 + GAT_N*8 + GAT_N*128)

// -------------------------------------------------------------------
// K0: fold node_w/node_b/att_w into W_alpha[128][16], b_alpha[16].
// Column h2 in [0,8)  -> src head h2   (uses node_w cols h*32 + 0..15)
// Column h2 in [8,16) -> dst head h2-8 (uses node_w cols h*32 + 16..31)
__global__ void gat_fold_weights(const float* __restrict__ node_w,
                                 const float* __restrict__ node_b,
                                 const float* __restrict__ att_w,
                                 float* __restrict__ Wa,
                                 float* __restrict__ ba) {
    int t = blockIdx.x * blockDim.x + threadIdx.x;
    if (t >= 128 * 16) return;
    int k  = t >> 4;
    int h2 = t & 15;
    int h   = h2 & 7;
    int off = (h2 >> 3) * 16;                 // 0 = src slice, 16 = dst slice
    const float* av   = att_w + h * 32 + off; // att_w is (8, 32) row-major
    const float* wrow = node_w + k * 256 + h * 32 + off;
    float s = 0.f;
#pragma unroll
    for (int d = 0; d < 16; ++d) s += wrow[d] * av[d];
    Wa[k * 16 + h2] = s;
    if (k == 0) {
        const float* brow = node_b + h * 32 + off;
        float sb = 0.f;
#pragma unroll
        for (int d = 0; d < 16; ++d) sb += brow[d] * av[d];
        ba[h2] = sb;
    }
}

// -------------------------------------------------------------------
// K1: alpha = X (N x 128) @ W_alpha (128 x 16) + b_alpha, via f32 WMMA.
// One wave32 per 16-row tile; 32 x V_WMMA_F32_16X16X4_F32 per tile.
__global__ void gat_alpha_gemm(const float* __restrict__ x,
                               const float* __restrict__ Wa,
                               const float* __restrict__ ba,
                               float* __restrict__ alpha,
                               int nTiles) {
    int wave = (blockIdx.x * blockDim.x + threadIdx.x) >> 5;
    int lane = threadIdx.x & 31;
    if (wave >= nTiles) return;                 // wave-uniform: EXEC stays all-1s
    const int row0  = wave * 16;
    const int m     = lane & 15;                // A: row M / B,C,D: col N
    const int khalf = (lane >> 4) ? 2 : 0;      // hi half-wave owns K=k0+2,k0+3
    const int hiM   = (lane >> 4) ? 8 : 0;

    v8f acc = {};
#pragma unroll
    for (int k0 = 0; k0 < 128; k0 += 4) {
        v2f a, b;
        const float* ap = x + (size_t)(row0 + m) * 128 + k0 + khalf;
        a.x = ap[0];                            // A[M][k0+khalf]
        a.y = ap[1];                            // A[M][k0+khalf+1]
        const float* bp = Wa + (k0 + khalf) * 16 + m;
        b.x = bp[0];                            // B[k0+khalf][N]
        b.y = bp[16];                           // B[k0+khalf+1][N]
        acc = __builtin_amdgcn_wmma_f32_16x16x4_f32(
            false, a, false, b, (short)0, acc, false, false);
    }
    const float bias = ba[m];
#pragma unroll
    for (int i = 0; i < 8; ++i) {
        int M = i + hiM;                        // D: VGPR i holds row M, col N=m
        alpha[(size_t)(row0 + M) * 16 + m] = acc[i] + bias;
    }
}

// -------------------------------------------------------------------
// K2: per (edge, head): leaky-ReLU score, store, segment-max via int-bit
// atomicMax (segmax init = 0.0f bits; clamp-to->=0 comes for free).
__global__ void gat_edge_scores(const long long* __restrict__ ei,
                                const float* __restrict__ alpha,
                                float* __restrict__ scores,
                                int* __restrict__ segmax_i) {
    long long t = (long long)blockIdx.x * 256 + threadIdx.x;
    if (t >= (long long)GAT_E * 8) return;
    long long e = t >> 3;
    int h = (int)(t & 7);
    long long r = ei[e];
    long long c = ei[GAT_E + e];
    float s = alpha[r * 16 + h] + alpha[c * 16 + 8 + h];
    s = (s >= 0.f) ? s : 0.2f * s;
    scores[t] = s;
    atomicMax(&segmax_i[c * 8 + h], __float_as_int(s));
}

// -------------------------------------------------------------------
// K3: per (edge, head): exp(s - segmax[col]), store back, segment-sum.
__global__ void gat_edge_exp(const long long* __restrict__ ei,
                             float* __restrict__ scores,
                             const float* __restrict__ segmax,
                             float* __restrict__ segsum) {
    long long t = (long long)blockIdx.x * 256 + threadIdx.x;
    if (t >= (long long)GAT_E * 8) return;
    long long e = t >> 3;
    int h = (int)(t & 7);
    long long c = ei[GAT_E + e];
    float es = __expf(scores[t] - segmax[c * 8 + h]);
    scores[t] = es;                               // reuse buffer as exp_s
    atomicAdd(&segsum[c * 8 + h], es);
}

// -------------------------------------------------------------------
// K4: one wave32 per edge. Lane l covers channels [4l, 4l+4) -> head l/4.
// float4 gather of x[row] (L2-resident), scaled scatter-add into agg[col].
__global__ void gat_edge_agg(const long long* __restrict__ ei,
                             const float* __restrict__ x,
                             const float* __restrict__ exps,
                             const float* __restrict__ segsum,
                             float* __restrict__ agg) {
    long long gid = (long long)blockIdx.x * 256 + threadIdx.x;
    long long e = gid >> 5;
    if (e >= (long long)GAT_E) return;
    int l = threadIdx.x & 31;
    long long r = ei[e];
    long long c = ei[GAT_E + e];
    int h = l >> 2;
    float attn = exps[e * 8 + h] / (segsum[c * 8 + h] + 1e-10f);
    const float4* xr = (const float4*)(x + r * 128);
    float4 xv = xr[l];
    float* a = agg + c * 128 + l * 4;
    atomicAdd(a + 0, xv.x * attn);
    atomicAdd(a + 1, xv.y * attn);
    atomicAdd(a + 2, xv.z * attn);
    atomicAdd(a + 3, xv.w * attn);
}

// -------------------------------------------------------------------
// K5: out = agg (N x 128) @ out_w (128 x 128) + out_b, via f32 WMMA.
// One wave owns a 16 x 128 row-stripe: A-frag reused across 8 column tiles.
__global__ void gat_out_gemm(const float* __restrict__ agg,
                             const float* __restrict__ W,
                             const float* __restrict__ bias,
                             float* __restrict__ out,
                             int nTiles) {
    int wave = (blockIdx.x * blockDim.x + threadIdx.x) >> 5;
    int lane = threadIdx.x & 31;
    if (wave >= nTiles) return;
    const int row0  = wave * 16;
    const int m     = lane & 15;
    const int khalf = (lane >> 4) ? 2 : 0;
    const int hiM   = (lane >> 4) ? 8 : 0;

    v8f acc[8];
#pragma unroll
    for (int t = 0; t < 8; ++t) acc[t] = (v8f){};

#pragma unroll 4
    for (int k0 = 0; k0 < 128; k0 += 4) {
        v2f a;
        const float* ap = agg + (size_t)(row0 + m) * 128 + k0 + khalf;
        a.x = ap[0];
        a.y = ap[1];
        const float* bp = W + (size_t)(k0 + khalf) * 128 + m;
#pragma unroll
        for (int t = 0; t < 8; ++t) {
            v2f b;
            b.x = bp[t * 16];                   // B[k0+khalf][t*16+m]
            b.y = bp[t * 16 + 128];             // B[k0+khalf+1][t*16+m]
            acc[t] = __builtin_amdgcn_wmma_f32_16x16x4_f32(
                false, a, false, b, (short)0, acc[t], false, false);
        }
    }
#pragma unroll
    for (int t = 0; t < 8; ++t) {
        float bv = bias[t * 16 + m];
#pragma unroll
        for (int i = 0; i < 8; ++i) {
            int M = i + hiM;
            out[(size_t)(row0 + M) * 128 + t * 16 + m] = acc[t][i] + bv;
        }
    }
}

// -------------------------------------------------------------------
extern "C" void kernel_launch(void* const* d_in, const int* in_sizes, int n_in,
                              void* d_out, int out_size, void* d_ws, size_t ws_size,
                              hipStream_t stream) {
    const float*     x      = (const float*)d_in[0];      // (N, 128)
    const long long* ei     = (const long long*)d_in[1];  // (2, E) int64
    const float*     node_w = (const float*)d_in[2];      // (128, 256)
    const float*     node_b = (const float*)d_in[3];      // (256,)
    const float*     att_w  = (const float*)d_in[4];      // (8, 32)
    const float*     out_w  = (const float*)d_in[5];      // (128, 128)
    const float*     out_b  = (const float*)d_in[6];      // (128,)
    float*           out    = (float*)d_out;              // (N, 128)

    float* ws      = (float*)d_ws;
    float* Wa      = ws + OFF_WALPHA;
    float* ba      = ws + OFF_BALPHA;
    float* alpha   = ws + OFF_ALPHA;
    float* scores  = ws + OFF_SCORES;
    float* segmax  = ws + OFF_SEGMAX;
    float* segsum  = ws + OFF_SEGSUM;
    float* agg     = ws + OFF_AGG;

    // Zero segmax (0.0f == bit pattern 0 -> doubles as the >=0 clamp),
    // segsum, and agg in one contiguous async memset (graph-capturable).
    hipMemsetAsync((void*)segmax, 0, (size_t)ZERO_FLOATS * sizeof(float), stream);

    // K0: fold weights (2048 threads).
    gat_fold_weights<<<8, 256, 0, stream>>>(node_w, node_b, att_w, Wa, ba);

    // K1: alpha GEMM. 3125 wave-tiles, 8 waves per 256-thread block.
    {
        int nTiles = GAT_N / 16;                       // 3125
        int blocks = (nTiles + 7) / 8;
        gat_alpha_gemm<<<blocks, 256, 0, stream>>>(x, Wa, ba, alpha, nTiles);
    }

    // K2/K3: per-(edge,head) phases, E*8 = 6.4M threads.
    {
        int blocks = (GAT_E * 8 + 255) / 256;          // 25000
        gat_edge_scores<<<blocks, 256, 0, stream>>>(ei, alpha, scores, (int*)segmax);
        gat_edge_exp<<<blocks, 256, 0, stream>>>(ei, scores, segmax, segsum);
    }

    // K4: one wave per edge (8 edges per 256-thread block).
    {
        int blocks = (GAT_E + 7) / 8;                  // 100000
        gat_edge_agg<<<blocks, 256, 0, stream>>>(ei, x, scores, segsum, agg);
    }

    // K5: output GEMM. 3125 row-stripe waves.
    {
        int nTiles = GAT_N / 16;
        int blocks = (nTiles + 7) / 8;
        gat_out_gemm<<<blocks, 256, 0, stream>>>(agg, out_w, out_b, out, nTiles);
    }
}